// SSIMLoss_16045997817858
// MI455X (gfx1250) — compile-verified
//
#include <hip/hip_runtime.h>
#include <math.h>

typedef float v2f __attribute__((ext_vector_type(2)));
typedef float v8f __attribute__((ext_vector_type(8)));

#define WMMA4(A, B, C) \
    __builtin_amdgcn_wmma_f32_16x16x4_f32(false, (A), false, (B), (short)0, (C), false, false)

#define IMG 512
#define PLANES 48                 // 16 batches * 3 channels
#define TILES_PER_PLANE 1024      // (512/16)^2
#define TOTAL_TILES (PLANES * TILES_PER_PLANE)   // 49152
#define WAVES_PER_BLOCK 4
#define NUM_BLOCKS (TOTAL_TILES / WAVES_PER_BLOCK)  // 12288
#define N_ELEMS 12582912.0f       // 16*3*512*512

// ---------------------------------------------------------------------------
// Main fused SSIM kernel: per wave, one 16x16 output tile.
//   Pass 1 (vertical blur):  V(16x32) = G(16x28) x In(28x32)   [WMMA, K=4 steps]
//   LDS transpose stage, then
//   Pass 2 (horizontal blur): Out(16x16) = V(16x28) x Gt(28x16) [WMMA]
// Five fields: x, y, x^2, y^2, x*y  ->  SSIM map -> block partial sum.
// ---------------------------------------------------------------------------
__global__ __launch_bounds__(WAVES_PER_BLOCK * 32)
void ssim_tile_kernel(const float* __restrict__ x,
                      const float* __restrict__ y,
                      float* __restrict__ partial) {
    __shared__ float lds[WAVES_PER_BLOCK * 5 * 16 * 32];   // 40 KB
    __shared__ float wsum[WAVES_PER_BLOCK];

    const int lane = threadIdx.x & 31;
    const int wave = threadIdx.x >> 5;
    const int l16  = lane & 15;       // row (A) or col (B/C) index
    const int up   = lane >> 4;       // lane-half selector

    const int tile  = blockIdx.x * WAVES_PER_BLOCK + wave;
    const int plane = tile >> 10;
    const int twi   = tile & 1023;
    const int r0    = (twi >> 5) << 4;
    const int c0    = (twi & 31) << 4;

    const float* xp = x + (size_t)plane * (IMG * IMG);
    const float* yp = y + (size_t)plane * (IMG * IMG);

    // ---- Gaussian taps g[0..10], sigma=1.5, normalized; 0 outside band ----
    const float inv2s2 = 1.0f / 4.5f;
    float S = 0.0f;
#pragma unroll
    for (int k = 0; k < 11; ++k) {
        float d = (float)(k - 5);
        S += __expf(-d * d * inv2s2);
    }
    const float invS = 1.0f / S;
    auto gAt = [&](int d) -> float {
        if (d < 0 || d > 10) return 0.0f;
        float t = (float)(d - 5);
        return __expf(-t * t * inv2s2) * invS;
    };

    // Band-matrix constants. Identity: A-layout of G and B-layout of Gt give
    // the same per-lane values g(kb - l16), so one table serves both passes.
    v2f gw[7];
#pragma unroll
    for (int k = 0; k < 7; ++k) {
        int kb = 4 * k + 2 * up;
        gw[k].x = gAt(kb - l16);
        gw[k].y = gAt(kb + 1 - l16);
    }

    // ---------------- Pass 1: vertical blur via WMMA ----------------
    v8f acc[5][2] = {};   // fields {x,y,xx,yy,xy} x two 16-col chunks
#pragma unroll
    for (int k = 0; k < 7; ++k) {
        const int ri = r0 - 5 + 4 * k + 2 * up;   // rows ri, ri+1 of this lane-half
#pragma unroll
        for (int n = 0; n < 2; ++n) {
            const int ci  = c0 - 5 + 16 * n + l16;
            const int cc  = ci < 0 ? 0 : (ci > IMG - 1 ? IMG - 1 : ci);
            const int ra  = ri < 0 ? 0 : (ri > IMG - 1 ? IMG - 1 : ri);
            const int rb  = (ri + 1) < 0 ? 0 : ((ri + 1) > IMG - 1 ? IMG - 1 : ri + 1);
            const bool cok = (ci >= 0) && (ci < IMG);
            const bool aok = cok && (ri >= 0) && (ri < IMG);
            const bool bok = cok && (ri + 1 >= 0) && (ri + 1 < IMG);
            // Clamp+select: loads always in-bounds, EXEC never diverges.
            float xa = xp[(size_t)ra * IMG + cc]; xa = aok ? xa : 0.0f;
            float xb = xp[(size_t)rb * IMG + cc]; xb = bok ? xb : 0.0f;
            float ya = yp[(size_t)ra * IMG + cc]; ya = aok ? ya : 0.0f;
            float yb = yp[(size_t)rb * IMG + cc]; yb = bok ? yb : 0.0f;
            v2f bx = {xa, xb}, by = {ya, yb};
            v2f bxx = bx * bx, byy = by * by, bxy = bx * by;
            acc[0][n] = WMMA4(gw[k], bx,  acc[0][n]);
            acc[1][n] = WMMA4(gw[k], by,  acc[1][n]);
            acc[2][n] = WMMA4(gw[k], bxx, acc[2][n]);
            acc[3][n] = WMMA4(gw[k], byy, acc[3][n]);
            acc[4][n] = WMMA4(gw[k], bxy, acc[4][n]);
        }
    }

    // ---------------- Stage V through LDS (D-layout -> row-major) ----------
    float* V = &lds[wave * (5 * 16 * 32)];
#pragma unroll
    for (int f = 0; f < 5; ++f)
#pragma unroll
        for (int n = 0; n < 2; ++n)
#pragma unroll
            for (int v = 0; v < 8; ++v) {
                int rr = v + 8 * up;
                V[f * 512 + rr * 32 + 16 * n + l16] = acc[f][n][v];
            }
    __syncthreads();

    // ---------------- Pass 2: horizontal blur via WMMA ----------------
    v8f out[5] = {};
#pragma unroll
    for (int k = 0; k < 7; ++k) {
        const int kb = 4 * k + 2 * up;
#pragma unroll
        for (int f = 0; f < 5; ++f) {
            // A-layout read: row=l16, K=(kb,kb+1); 8B-aligned -> ds_load_b64
            v2f a = *(const v2f*)&V[f * 512 + l16 * 32 + kb];
            out[f] = WMMA4(a, gw[k], out[f]);
        }
    }

    // ---------------- SSIM map + reduction ----------------
    const float C1 = 0.0001f, C2 = 0.0009f;
    float ssum = 0.0f;
#pragma unroll
    for (int i = 0; i < 8; ++i) {
        float mx  = out[0][i], my  = out[1][i];
        float exx = out[2][i], eyy = out[3][i], exy = out[4][i];
        float mx2 = mx * mx, my2 = my * my, mxy = mx * my;
        float sx2 = exx - mx2, sy2 = eyy - my2, sxy = exy - mxy;
        float num = (2.0f * mxy + C1) * (2.0f * sxy + C2);
        float den = (mx2 + my2 + C1) * (sx2 + sy2 + C2) + 1e-12f;
        ssum += num / den;
    }
#pragma unroll
    for (int off = 16; off > 0; off >>= 1)
        ssum += __shfl_xor(ssum, off, 32);
    if (lane == 0) wsum[wave] = ssum;
    __syncthreads();
    if (threadIdx.x == 0) {
        float b = 0.0f;
#pragma unroll
        for (int w = 0; w < WAVES_PER_BLOCK; ++w) b += wsum[w];
        partial[blockIdx.x] = b;   // deterministic: no atomics
    }
}

// Deterministic final reduction of NUM_BLOCKS partials -> scalar loss.
__global__ __launch_bounds__(256)
void ssim_finalize_kernel(const float* __restrict__ partial, float* __restrict__ out) {
    __shared__ float red[256];
    float s = 0.0f;
    for (int i = threadIdx.x; i < NUM_BLOCKS; i += 256) s += partial[i];
    red[threadIdx.x] = s;
    __syncthreads();
    for (int step = 128; step > 0; step >>= 1) {
        if (threadIdx.x < step) red[threadIdx.x] += red[threadIdx.x + step];
        __syncthreads();
    }
    if (threadIdx.x == 0) out[0] = 1.0f - red[0] * (1.0f / N_ELEMS);
}

extern "C" void kernel_launch(void* const* d_in, const int* in_sizes, int n_in,
                              void* d_out, int out_size, void* d_ws, size_t ws_size,
                              hipStream_t stream) {
    const float* x = (const float*)d_in[0];
    const float* y = (const float*)d_in[1];
    // d_in[2] (window) ignored: taps are regenerated exactly in-kernel.
    float* partial = (float*)d_ws;           // NUM_BLOCKS floats (48 KB)
    float* out     = (float*)d_out;

    ssim_tile_kernel<<<NUM_BLOCKS, WAVES_PER_BLOCK * 32, 0, stream>>>(x, y, partial);
    ssim_finalize_kernel<<<1, 256, 0, stream>>>(partial, out);
}